// MotifVector_24335284699142
// MI455X (gfx1250) — compile-verified
//
#include <hip/hip_runtime.h>

typedef __attribute__((ext_vector_type(2))) float v2f;
typedef __attribute__((ext_vector_type(4))) float v4f;
typedef __attribute__((ext_vector_type(8))) float v8f;

#define N_HIDDEN   512
#define N_MOTIF    160
#define NCT        10      // 10 column tiles of 16 motifs == 10 classes
#define B_TOTAL    65536
#define ROWS_PER_WG 128    // 8 waves * 16 rows
#define NUM_WG     (B_TOTAL / ROWS_PER_WG)   // 512
#define EPS_D      1e-4f

// ---------------- kernel 0: mm[j] = sum_k M[j][k]^2 (160 values) ----------------
__global__ __launch_bounds__(256) void motif_mm_kernel(const float* __restrict__ M,
                                                       float* __restrict__ mm) {
  int t = threadIdx.x;
  if (t < N_MOTIF) {
    const v4f* p = reinterpret_cast<const v4f*>(M + (size_t)t * N_HIDDEN);
    float s = 0.f;
#pragma unroll 4
    for (int k = 0; k < N_HIDDEN / 4; ++k) {
      v4f v = p[k];
      s = __builtin_fmaf(v.x, v.x, s);
      s = __builtin_fmaf(v.y, v.y, s);
      s = __builtin_fmaf(v.z, v.z, s);
      s = __builtin_fmaf(v.w, v.w, s);
    }
    mm[t] = s;
  }
}

// ---------------- kernel 1: fused GEMM (f32 WMMA) + transform + row reduce ----------------
// Each wave: 16 rows of z, all 160 motifs, K=512 via V_WMMA_F32_16X16X4_F32.
//
// K-remap trick: the contraction is order-agnostic, so we assign each lane a
// contiguous 4-float chunk (b128 load) at phys offset k + 4*half and feed TWO
// WMMAs from it ({x,y} then {z,w}). A and B use the identical (lane,slot)->k
// mapping, so every physical k is paired correctly and covered exactly once.
//   C/D (16x16): VGPR i, lane L -> row i+8*(L>>4), col (L&15).
__global__ __launch_bounds__(256) void motif_loss_kernel(const float* __restrict__ z,
                                                         const float* __restrict__ M,
                                                         const int*   __restrict__ y,
                                                         const float* __restrict__ mm,
                                                         float* __restrict__ partial) {
  const int lane = threadIdx.x & 31;
  const int wave = threadIdx.x >> 5;
  const int half = lane >> 4;     // k-chunk selector
  const int lrow = lane & 15;     // fragment row
  const int rowbase = blockIdx.x * ROWS_PER_WG + wave * 16;

  const float* pA = z + (size_t)(rowbase + lrow) * N_HIDDEN + 4 * half;
  const float* pB = M + (size_t)lrow * N_HIDDEN + 4 * half;

  v8f acc[NCT];
#pragma unroll
  for (int ct = 0; ct < NCT; ++ct) acc[ct] = (v8f){0.f,0.f,0.f,0.f,0.f,0.f,0.f,0.f};

  float zzp = 0.f;   // partial ||z_row||^2 (this lane's k-chunks)

#pragma unroll 2
  for (int k = 0; k < N_HIDDEN; k += 8) {
    // A fragment: 16B/lane, streamed once -> non-temporal
    v4f a = __builtin_nontemporal_load(reinterpret_cast<const v4f*>(pA + k));
    zzp = __builtin_fmaf(a.x, a.x, zzp);
    zzp = __builtin_fmaf(a.y, a.y, zzp);
    zzp = __builtin_fmaf(a.z, a.z, zzp);
    zzp = __builtin_fmaf(a.w, a.w, zzp);
    v2f a0 = __builtin_shufflevector(a, a, 0, 1);
    v2f a1 = __builtin_shufflevector(a, a, 2, 3);

    // B fragments: 10 x b128 (hot in L0/L2; M is only 320 KB)
    v4f bf[NCT];
#pragma unroll
    for (int ct = 0; ct < NCT; ++ct)
      bf[ct] = *reinterpret_cast<const v4f*>(pB + (size_t)ct * 16 * N_HIDDEN + k);

    // two WMMA sweeps; same-accumulator WMMAs stay 10 instructions apart
#pragma unroll
    for (int ct = 0; ct < NCT; ++ct) {
      v2f b0 = __builtin_shufflevector(bf[ct], bf[ct], 0, 1);
      acc[ct] = __builtin_amdgcn_wmma_f32_16x16x4_f32(
          false, a0, false, b0, (short)0, acc[ct], false, false);
    }
#pragma unroll
    for (int ct = 0; ct < NCT; ++ct) {
      v2f b1 = __builtin_shufflevector(bf[ct], bf[ct], 2, 3);
      acc[ct] = __builtin_amdgcn_wmma_f32_16x16x4_f32(
          false, a1, false, b1, (short)0, acc[ct], false, false);
    }
  }

  // full ||z_row||^2 for row 'lrow' (combine the two k-chunk halves)
  float zzfull = zzp + __shfl_xor(zzp, 16, 32);

  // per-lane: zz and class label for the 8 C-rows this lane holds
  float zrow[8];
  int   ycls[8];
#pragma unroll
  for (int i = 0; i < 8; ++i) {
    zrow[i] = __shfl(zzfull, 8 * half + i, 32);
    ycls[i] = y[rowbase + 8 * half + i];
  }

  float pos[8], neg[8];
#pragma unroll
  for (int i = 0; i < 8; ++i) { pos[i] = 0.f; neg[i] = 0.f; }

#pragma unroll
  for (int ct = 0; ct < NCT; ++ct) {
    float mms = mm[ct * 16 + lrow];
#pragma unroll
    for (int i = 0; i < 8; ++i) {
      float xp   = acc[ct][i];
      float dist = zrow[i] + mms - 2.f * xp;
      // exp(log((d+1)/(d+eps)) / 0.2) == ((d+1)/(d+eps))^5  exactly
      float r  = (dist + 1.f) / (dist + EPS_D);
      float r2 = r * r;
      float r4 = r2 * r2;
      float s5 = r4 * r;
      bool  p  = (ct == ycls[i]);
      pos[i] += p ? s5 : 0.f;
      neg[i] += p ? 0.f : s5;
    }
  }

  // reduce across the 16 columns (lanes within a half-wave)
#pragma unroll
  for (int off = 1; off <= 8; off <<= 1) {
#pragma unroll
    for (int i = 0; i < 8; ++i) {
      pos[i] += __shfl_xor(pos[i], off, 32);
      neg[i] += __shfl_xor(neg[i], off, 32);
    }
  }

  // per-row loss: -log(pos/(pos+neg)) = log((pos+neg)/pos); sum this lane's 8 rows
  float s = 0.f;
#pragma unroll
  for (int i = 0; i < 8; ++i) s += logf((pos[i] + neg[i]) / pos[i]);
  // combine the two halves (each half's value is replicated across its 16 lanes)
  s += __shfl_xor(s, 16, 32);

  __shared__ float wsum[8];
  if (lane == 0) wsum[wave] = s;
  __syncthreads();
  if (threadIdx.x == 0) {
    float t = 0.f;
#pragma unroll
    for (int w = 0; w < 8; ++w) t += wsum[w];   // fixed order -> deterministic
    partial[blockIdx.x] = t;
  }
}

// ---------------- kernel 2: deterministic final reduction + mean ----------------
__global__ __launch_bounds__(256) void motif_reduce_kernel(const float* __restrict__ partial,
                                                           float* __restrict__ out) {
  __shared__ float lds[256];
  int t = threadIdx.x;
  lds[t] = partial[t] + partial[t + 256];
  __syncthreads();
  for (int s = 128; s > 0; s >>= 1) {
    if (t < s) lds[t] += lds[t + s];
    __syncthreads();
  }
  if (t == 0) out[0] = lds[0] * (1.0f / (float)B_TOTAL);
}

extern "C" void kernel_launch(void* const* d_in, const int* in_sizes, int n_in,
                              void* d_out, int out_size, void* d_ws, size_t ws_size,
                              hipStream_t stream) {
  const float* z = (const float*)d_in[0];   // [65536, 512]
  const float* M = (const float*)d_in[1];   // [160, 512]
  const int*   y = (const int*)d_in[2];     // [65536]
  float* out = (float*)d_out;

  float* mm      = (float*)d_ws;            // 160 floats (padded to 256)
  float* partial = mm + 256;                // 512 floats

  motif_mm_kernel<<<1, 256, 0, stream>>>(M, mm);
  motif_loss_kernel<<<NUM_WG, 256, 0, stream>>>(z, M, y, mm, partial);
  motif_reduce_kernel<<<1, 256, 0, stream>>>(partial, out);
}